// PCKhLoss_55731495633067
// MI455X (gfx1250) — compile-verified
//
#include <hip/hip_runtime.h>
#include <math.h>

typedef __attribute__((ext_vector_type(2))) float v2f;
typedef __attribute__((ext_vector_type(8))) float v8f;

#define B_   256
#define J_   17
#define H_   64
#define W_   64
#define HW_  4096            // H_*W_
#define BSTRIDE (J_ * HW_)   // 69632 floats between batches

// ---------------------------------------------------------------------------
// Kernel 1: S[tensor][j][hw] = sum_b exp(x[b][j][hw]) via WMMA f32 16x16x4
// with A = ones. D[m,n] = sum_k B[k,n]; all rows of D identical, so any C VGPR
// holds the column sum for n = lane%16. Full f32 math — bit-compatible class
// of accuracy with the reference (only summation order differs).
// One wave per 16-wide hw tile; 64 chained WMMAs cover all 256 batches.
// ---------------------------------------------------------------------------
__global__ void pckh_batchsum_exp_wmma(const float* __restrict__ out,
                                       const float* __restrict__ tgt,
                                       float* __restrict__ S) {
    const int wavesPerBlock = blockDim.x >> 5;
    const int waveId = blockIdx.x * wavesPerBlock + ((int)threadIdx.x >> 5);
    const int lane   = (int)threadIdx.x & 31;

    const int tilesPerTensor = J_ * (HW_ / 16);      // 4352
    const int tensor = waveId / tilesPerTensor;
    const int rem    = waveId - tensor * tilesPerTensor;
    const int j      = rem / (HW_ / 16);
    const int tile   = rem - j * (HW_ / 16);

    const int n    = lane & 15;       // output column / hw offset within tile
    const int half = lane >> 4;       // which K pair this lane supplies

    const float* src  = (tensor == 0) ? out : tgt;
    const float* base = src + (size_t)j * HW_ + (size_t)tile * 16 + n;

    v2f a; a.x = 1.0f; a.y = 1.0f;    // ones A-matrix (layout-invariant)
    v8f c = {};

    #pragma unroll 4
    for (int b0 = 0; b0 < B_; b0 += 4) {
        const int k = b0 + half * 2;
        v2f bm;
        bm.x = expf(base[(size_t)k * BSTRIDE]);
        bm.y = expf(base[(size_t)(k + 1) * BSTRIDE]);
        c = __builtin_amdgcn_wmma_f32_16x16x4_f32(
                /*neg_a=*/false, a, /*neg_b=*/false, bm,
                /*c_mod=*/(short)0, c, /*reuse_a=*/false, /*reuse_b=*/false);
    }

    if (lane < 16) {
        S[(size_t)tensor * (J_ * HW_) + (size_t)j * HW_ + (size_t)tile * 16 + n] = c[0];
    }
}

// ---------------------------------------------------------------------------
// Kernel 2: per (tensor, b, j): argmax over hw of exp(x)/S  ->  (px, py).
// First-occurrence tie-break to match jnp.argmax semantics.
// ---------------------------------------------------------------------------
__global__ void pckh_argmax_preds(const float* __restrict__ out,
                                  const float* __restrict__ tgt,
                                  const float* __restrict__ S,
                                  float* __restrict__ preds) {
    __shared__ float sv[256];
    __shared__ int   si[256];

    const int tensor = blockIdx.x / (B_ * J_);
    const int rem    = blockIdx.x % (B_ * J_);   // = b*J_ + j
    const int b      = rem / J_;
    const int j      = rem % J_;

    const float* src = ((tensor == 0) ? out : tgt) + ((size_t)b * J_ + j) * HW_;
    const float* Sj  = S + (size_t)tensor * (J_ * HW_) + (size_t)j * HW_;

    const int tid = (int)threadIdx.x;
    float best = -1.0f;
    int   bidx = 0x7fffffff;
    for (int i = tid; i < HW_; i += 256) {
        const float v = expf(src[i]) / Sj[i];
        if (v > best) { best = v; bidx = i; }   // strict > keeps earliest index
    }
    sv[tid] = best; si[tid] = bidx;
    __syncthreads();
    for (int off = 128; off > 0; off >>= 1) {
        if (tid < off) {
            const float ov = sv[tid + off];
            const int   oi = si[tid + off];
            if (ov > sv[tid] || (ov == sv[tid] && oi < si[tid])) {
                sv[tid] = ov; si[tid] = oi;
            }
        }
        __syncthreads();
    }
    if (tid == 0) {
        const int idx = si[0];
        const float m = (sv[0] > 0.0f) ? 1.0f : 0.0f;  // softmax > 0 always
        const float px = (float)(idx % W_);
        const float py = (float)(idx / W_);
        float* p = preds + ((size_t)tensor * B_ * J_ + rem) * 2;
        p[0] = px * m;
        p[1] = py * m;
    }
}

// ---------------------------------------------------------------------------
// Kernel 3: distances, per-joint accuracy, average. One block, thread = batch.
// norm = [H,W]/10 = 6.4 in both components; thr = 0.5.
// ---------------------------------------------------------------------------
__global__ void pckh_final(const float* __restrict__ preds,
                           float* __restrict__ outv) {
    __shared__ int s_cal[256];
    __shared__ int s_bel[256];

    const int t = (int)threadIdx.x;                 // batch index
    const float* po = preds;                        // output preds
    const float* pt = preds + (size_t)B_ * J_ * 2;  // target preds

    float accsum = 0.0f;
    int   acccnt = 0;

    for (int j = 0; j < J_; ++j) {
        const int base = (t * J_ + j) * 2;
        const float pox = po[base], poy = po[base + 1];
        const float ptx = pt[base], pty = pt[base + 1];
        const bool valid = (ptx > 1.0f) && (pty > 1.0f);
        const float dx = (pox - ptx) / 6.4f;
        const float dy = (poy - pty) / 6.4f;
        const float d  = sqrtf(dx * dx + dy * dy);
        s_cal[t] = valid ? 1 : 0;
        s_bel[t] = (valid && d < 0.5f) ? 1 : 0;
        __syncthreads();
        for (int off = 128; off > 0; off >>= 1) {
            if (t < off) { s_cal[t] += s_cal[t + off]; s_bel[t] += s_bel[t + off]; }
            __syncthreads();
        }
        if (t == 0) {
            const int num = s_cal[0], below = s_bel[0];
            const float aj = (num > 0) ? (float)below / (float)num : -1.0f;
            outv[1 + j] = aj;
            if (aj >= 0.0f) { accsum += aj; ++acccnt; }
        }
        __syncthreads();
    }
    if (t == 0) {
        outv[0] = (acccnt > 0) ? accsum / (float)acccnt : 0.0f;
    }
}

extern "C" void kernel_launch(void* const* d_in, const int* in_sizes, int n_in,
                              void* d_out, int out_size, void* d_ws, size_t ws_size,
                              hipStream_t stream) {
    (void)in_sizes; (void)n_in; (void)out_size; (void)ws_size;
    const float* out = (const float*)d_in[0];
    const float* tgt = (const float*)d_in[1];

    float* S     = (float*)d_ws;                 // 2 * 17 * 4096 floats
    float* preds = S + 2 * (size_t)J_ * HW_;     // 2 * 256 * 17 * 2 floats

    // Kernel 1: 8704 waves total, 8 waves/block -> 1088 blocks (exact fit,
    // EXEC all-ones at every WMMA).
    const int totalWaves = 2 * J_ * (HW_ / 16);  // 8704
    pckh_batchsum_exp_wmma<<<totalWaves / 8, 256, 0, stream>>>(out, tgt, S);

    // Kernel 2: one block per (tensor, b, j).
    pckh_argmax_preds<<<2 * B_ * J_, 256, 0, stream>>>(out, tgt, S, preds);

    // Kernel 3: single block epilogue.
    pckh_final<<<1, 256, 0, stream>>>(preds, (float*)d_out);
}